// RoIHeads_87763361727040
// MI455X (gfx1250) — compile-verified
//
#include <hip/hip_runtime.h>
#include <hip/hip_bf16.h>
#include <stdint.h>

// ---------------------------------------------------------------------------
// Mask R-CNN RoIHeads postprocess for MI455X (gfx1250, wave32).
//  K_A: fused softmax + box decode -> class-major scores/boxes in ws
//  K_B: per-(image,class) NMS: async-to-LDS box staging, compaction,
//       bitonic sort, WMMA-built pairwise area-sum tile, greedy suppression
//  K_C: per-image bitonic top-K=100 over surviving candidates
//  K_D: mask channel gather + sigmoid
// ---------------------------------------------------------------------------

#define BQ    8
#define NP    1000
#define NC    81
#define CLS   (NC - 1)
#define NKEEP 100
#define MRES  28
#define NPAD  1024
#define MAXC  512      // per-class candidate cap (expected ~30)
#define WREG  64       // WMMA-precomputed pairwise region
#define CAP   4096     // per-image candidate cap (expected ~2300)

#define IMGW 1216.0f
#define IMGH 800.0f
#define XFORM_CLIP 4.135166556742356f   // log(1000/16)

typedef float v2f __attribute__((ext_vector_type(2)));
typedef float v8f __attribute__((ext_vector_type(8)));

// ---------------- ws layout (float offsets) ----------------
// scoresW : [B, CLS, NPAD]              = 655360
// boxesW  : [B, CLS, NPAD, 4]           = 2621440
// candBuf : [B, CAP, 8]                 = 262144
// gcnt    : 8 ints
// labels_i: 800 ints
#define WS_SCORES 0
#define WS_BOXES  655360
#define WS_CAND   3276800
#define WS_GCNT   3538944
#define WS_LAB    3538952

// ===========================================================================
// Kernel A: softmax over 81 classes + box decode, class-major output.
// One block (128 thr) per proposal row; thread c owns class c.
// ===========================================================================
__global__ void kA_softmax_decode(const float* __restrict__ logits,
                                  const float* __restrict__ reg,
                                  const float* __restrict__ props,
                                  float* __restrict__ scoresW,
                                  float* __restrict__ boxesW,
                                  int* __restrict__ gcnt) {
  const int row = blockIdx.x;            // 0..7999
  const int t   = threadIdx.x;           // 0..127
  const int b   = row / NP;
  const int n   = row % NP;
  if (row == 0 && t < BQ) gcnt[t] = 0;   // reset per-image candidate counters

  __shared__ float red[128];
  float l = (t < NC) ? logits[(size_t)row * NC + t] : -3.4e38f;
  red[t] = l; __syncthreads();
  for (int s = 64; s > 0; s >>= 1) {
    if (t < s) red[t] = fmaxf(red[t], red[t + s]);
    __syncthreads();
  }
  float mx = red[0]; __syncthreads();
  float e = (t < NC) ? __expf(l - mx) : 0.f;
  red[t] = e; __syncthreads();
  for (int s = 64; s > 0; s >>= 1) {
    if (t < s) red[t] += red[t + s];
    __syncthreads();
  }
  const float denom = red[0];

  if (t >= 1 && t < NC) {                // skip background class 0
    float sc = e / denom;
    float x1 = props[row * 4 + 0], y1 = props[row * 4 + 1];
    float x2 = props[row * 4 + 2], y2 = props[row * 4 + 3];
    float w = x2 - x1, h = y2 - y1;
    float cx = x1 + 0.5f * w, cy = y1 + 0.5f * h;
    const float* r4 = reg + (size_t)row * NC * 4 + t * 4;
    float dx = r4[0] * 0.1f;                          // /WX
    float dy = r4[1] * 0.1f;                          // /WY
    float dw = fminf(r4[2] * 0.2f, XFORM_CLIP);       // /WW
    float dh = fminf(r4[3] * 0.2f, XFORM_CLIP);       // /WH
    float pcx = dx * w + cx, pcy = dy * h + cy;
    float pw = __expf(dw) * w, ph = __expf(dh) * h;
    float bx1 = fminf(fmaxf(pcx - 0.5f * pw, 0.f), IMGW);
    float by1 = fminf(fmaxf(pcy - 0.5f * ph, 0.f), IMGH);
    float bx2 = fminf(fmaxf(pcx + 0.5f * pw, 0.f), IMGW);
    float by2 = fminf(fmaxf(pcy + 0.5f * ph, 0.f), IMGH);
    size_t ci = (size_t)(b * CLS + (t - 1));
    scoresW[ci * NPAD + n] = sc;
    float4* bp = (float4*)(boxesW + (ci * NPAD + n) * 4);  // 16B aligned
    *bp = make_float4(bx1, by1, bx2, by2);
  }
}

// ===========================================================================
// Kernel B: per-(image,class) NMS.  256 threads = 8 waves.
// Dynamic LDS layout (byte offsets, dynamic region assumed at LDS base 0
// so async-to-LDS offsets match pointer indexing):
//   0      boxesLds 1024*16 = 16384
//   16384  cscore   512*4
//   18432  cidx     512*4
//   20480  keep     512*4
//   22528  areas    512*4
//   24576  sumA     64*68*4 = 17408
//   41984  cnt      + pad
// ===========================================================================
#define SMEM_B 42112

__global__ void kB_nms(const float* __restrict__ scoresW,
                       const float* __restrict__ boxesW,
                       float* __restrict__ candBuf,
                       int* __restrict__ gcnt) {
  extern __shared__ char smem[];
  float* boxesLds = (float*)(smem);
  float* cscore   = (float*)(smem + 16384);
  int*   cidx     = (int*)  (smem + 18432);
  int*   keep     = (int*)  (smem + 20480);
  float* areas    = (float*)(smem + 22528);
  float* sumA     = (float*)(smem + 24576);   // [64][68]
  int*   cntp     = (int*)  (smem + 41984);

  const int t   = threadIdx.x;                // 0..255
  const int blk = blockIdx.x;                 // 0..639
  const int b   = blk / CLS;
  const int c   = blk % CLS;
  const size_t base = (size_t)(b * CLS + c) * NPAD;
  const float* gScores = scoresW + base;
  const float* gBoxes  = boxesW + base * 4;

  if (t == 0) *cntp = 0;

  // --- CDNA5 async copy: stage all 1000 boxes (16B each) into LDS ---
  for (int n = t; n < NP; n += 256) {
    unsigned lds_off = (unsigned)(n * 16);
    unsigned long long ga = (unsigned long long)(uintptr_t)(gBoxes + n * 4);
    asm volatile("global_load_async_to_lds_b128 %0, %1, off"
                 :: "v"(lds_off), "v"(ga) : "memory");
  }
  asm volatile("s_wait_asynccnt 0" ::: "memory");
  __syncthreads();

  // --- compact candidates with score > SCORE_THRESH ---
  for (int n = t; n < NP; n += 256) {
    float s = gScores[n];
    if (s > 0.05f) {
      int p = atomicAdd(cntp, 1);
      if (p < MAXC) { cscore[p] = s; cidx[p] = n; }
    }
  }
  __syncthreads();
  int cnt = *cntp; if (cnt > MAXC) cnt = MAXC;
  for (int r = t; r < MAXC; r += 256) {
    if (r >= cnt) { cscore[r] = -1e30f; cidx[r] = 0; }
  }
  __syncthreads();

  // --- bitonic sort MAXC records descending by score ---
  for (int k = 2; k <= MAXC; k <<= 1) {
    for (int j = k >> 1; j > 0; j >>= 1) {
      for (int i = t; i < MAXC; i += 256) {
        int ixj = i ^ j;
        if (ixj > i) {
          bool up = ((i & k) == 0);
          float s1 = cscore[i], s2 = cscore[ixj];
          if (up ? (s1 < s2) : (s1 > s2)) {
            cscore[i] = s2; cscore[ixj] = s1;
            int tmp = cidx[i]; cidx[i] = cidx[ixj]; cidx[ixj] = tmp;
          }
        }
      }
      __syncthreads();
    }
  }

  // --- areas + keep init ---
  for (int r = t; r < MAXC; r += 256) {
    const float* bp = boxesLds + cidx[r] * 4;
    areas[r] = (bp[2] - bp[0]) * (bp[3] - bp[1]);
    keep[r] = (r < cnt) ? 1 : 0;
  }
  __syncthreads();

  // --- WMMA: pairwise area-sum  sumA[i][j] = a[i] + a[j]  (64x64 region)
  //     rank-2 outer product: A row m = {a_i[m], 1, 0, 0};  B = {{1..1},{a_j},0,0}
  {
    const int wave = t >> 5, lane = t & 31;
    const bool lo  = lane < 16;
    const int  l16 = lane & 15;
    for (int tile = wave; tile < 16; tile += 8) {      // 4x4 tiles of 16x16
      const int ti = (tile >> 2) * 16;
      const int tj = (tile & 3) * 16;
      v2f a, bm;
      a.x  = lo ? areas[ti + l16] : 0.f;   // A: K=0 row (lanes 0-15), K=2 zero
      a.y  = lo ? 1.f : 0.f;               //    K=1 row,              K=3 zero
      bm.x = lo ? 1.f : 0.f;               // B: K=0 row = ones
      bm.y = lo ? areas[tj + l16] : 0.f;   //    K=1 row = a_j
      v8f acc = {};
      acc = __builtin_amdgcn_wmma_f32_16x16x4_f32(
          false, a, false, bm, (short)0, acc, false, false);
#pragma unroll
      for (int r8 = 0; r8 < 8; ++r8)
        sumA[(ti + r8 + (lo ? 0 : 8)) * 68 + (tj + l16)] = acc[r8];
    }
  }
  __syncthreads();

  // --- greedy NMS: sequential over i, parallel over j (2 per thread) ---
  const int jj0 = t, jj1 = t + 256;
  float j0b0 = boxesLds[cidx[jj0] * 4 + 0], j0b1 = boxesLds[cidx[jj0] * 4 + 1];
  float j0b2 = boxesLds[cidx[jj0] * 4 + 2], j0b3 = boxesLds[cidx[jj0] * 4 + 3];
  float j1b0 = boxesLds[cidx[jj1] * 4 + 0], j1b1 = boxesLds[cidx[jj1] * 4 + 1];
  float j1b2 = boxesLds[cidx[jj1] * 4 + 2], j1b3 = boxesLds[cidx[jj1] * 4 + 3];

  for (int i = 0; i < cnt; ++i) {
    __syncthreads();
    if (!keep[i]) continue;                       // uniform across block
    const float* ib = boxesLds + cidx[i] * 4;
    const float ix1 = ib[0], iy1 = ib[1], ix2 = ib[2], iy2 = ib[3];
    const float ia  = areas[i];
    if (jj0 > i && jj0 < cnt && keep[jj0]) {
      float iw = fmaxf(fminf(ix2, j0b2) - fmaxf(ix1, j0b0), 0.f);
      float ih = fmaxf(fminf(iy2, j0b3) - fmaxf(iy1, j0b1), 0.f);
      float inter = iw * ih;
      float sa = (i < WREG && jj0 < WREG) ? sumA[i * 68 + jj0] : (ia + areas[jj0]);
      if (inter > 0.5f * (sa - inter + 1e-9f)) keep[jj0] = 0;
    }
    if (jj1 > i && jj1 < cnt && keep[jj1]) {
      float iw = fmaxf(fminf(ix2, j1b2) - fmaxf(ix1, j1b0), 0.f);
      float ih = fmaxf(fminf(iy2, j1b3) - fmaxf(iy1, j1b1), 0.f);
      float inter = iw * ih;
      float sa = (i < WREG && jj1 < WREG) ? sumA[i * 68 + jj1] : (ia + areas[jj1]);
      if (inter > 0.5f * (sa - inter + 1e-9f)) keep[jj1] = 0;
    }
  }
  __syncthreads();

  // --- append survivors to per-image candidate list ---
  if (jj0 < cnt && keep[jj0]) {
    int slot = atomicAdd(&gcnt[b], 1);
    if (slot < CAP) {
      float* rec = candBuf + ((size_t)b * CAP + slot) * 8;
      rec[0] = cscore[jj0];
      rec[1] = j0b0; rec[2] = j0b1; rec[3] = j0b2; rec[4] = j0b3;
      rec[5] = (float)(c + 1); rec[6] = 0.f; rec[7] = 0.f;
    }
  }
  if (jj1 < cnt && keep[jj1]) {
    int slot = atomicAdd(&gcnt[b], 1);
    if (slot < CAP) {
      float* rec = candBuf + ((size_t)b * CAP + slot) * 8;
      rec[0] = cscore[jj1];
      rec[1] = j1b0; rec[2] = j1b1; rec[3] = j1b2; rec[4] = j1b3;
      rec[5] = (float)(c + 1); rec[6] = 0.f; rec[7] = 0.f;
    }
  }
}

// ===========================================================================
// Kernel C: per-image top-K=100 via bitonic sort of candidate scores.
// ===========================================================================
__global__ void kC_topk(const float* __restrict__ candBuf,
                        const int* __restrict__ gcnt,
                        float* __restrict__ out,
                        int* __restrict__ labels_i) {
  __shared__ float key[CAP];
  __shared__ int   kid[CAP];
  const int b = blockIdx.x, t = threadIdx.x;  // 256 threads
  int cnt = gcnt[b]; if (cnt > CAP) cnt = CAP;
  for (int i = t; i < CAP; i += 256) {
    key[i] = (i < cnt) ? candBuf[((size_t)b * CAP + i) * 8] : -1.f;
    kid[i] = i;
  }
  __syncthreads();
  for (int k = 2; k <= CAP; k <<= 1) {
    for (int j = k >> 1; j > 0; j >>= 1) {
      for (int i = t; i < CAP; i += 256) {
        int ixj = i ^ j;
        if (ixj > i) {
          bool up = ((i & k) == 0);
          float s1 = key[i], s2 = key[ixj];
          if (up ? (s1 < s2) : (s1 > s2)) {
            key[i] = s2; key[ixj] = s1;
            int tmp = kid[i]; kid[i] = kid[ixj]; kid[ixj] = tmp;
          }
        }
      }
      __syncthreads();
    }
  }
  if (t < NKEEP) {
    float s = key[t];
    bool valid = s > 0.f;
    const float* rec = candBuf + ((size_t)b * CAP + kid[t]) * 8;
    float bx1 = valid ? rec[1] : 0.f, by1 = valid ? rec[2] : 0.f;
    float bx2 = valid ? rec[3] : 0.f, by2 = valid ? rec[4] : 0.f;
    int lab = valid ? (int)rec[5] : 0;
    float* tb = out + ((size_t)b * NKEEP + t) * 4;
    tb[0] = bx1; tb[1] = by1; tb[2] = bx2; tb[3] = by2;
    out[3200 + b * NKEEP + t] = valid ? s : 0.f;   // top_s
    out[4000 + b * NKEEP + t] = (float)lab;        // labels
    labels_i[b * NKEEP + t] = lab;
  }
}

// ===========================================================================
// Kernel D: gather per-detection mask channel + sigmoid.
// ===========================================================================
__global__ void kD_mask(const float* __restrict__ mlog,
                        const int* __restrict__ labels_i,
                        float* __restrict__ out) {
  const int blk = blockIdx.x;          // b*100 + k, 0..799
  const int t   = threadIdx.x;         // 256
  const int lab = labels_i[blk];
  const float* src = mlog + ((size_t)blk * NC + lab) * (MRES * MRES);
  float* dst = out + 4800 + (size_t)blk * (MRES * MRES);
  for (int i = t; i < MRES * MRES; i += 256) {
    __builtin_prefetch(src + i + 256, 0, 0);       // global_prefetch_b8
    dst[i] = 1.f / (1.f + __expf(-src[i]));
  }
}

// ===========================================================================
extern "C" void kernel_launch(void* const* d_in, const int* in_sizes, int n_in,
                              void* d_out, int out_size, void* d_ws, size_t ws_size,
                              hipStream_t stream) {
  const float* logits = (const float*)d_in[0];   // [8000, 81]
  const float* reg    = (const float*)d_in[1];   // [8000, 324]
  const float* props  = (const float*)d_in[2];   // [8, 1000, 4]
  const float* mlog   = (const float*)d_in[3];   // [8, 100, 81, 28, 28]
  float* out = (float*)d_out;                    // 632000 floats

  float* wsf      = (float*)d_ws;
  float* scoresW  = wsf + WS_SCORES;
  float* boxesW   = wsf + WS_BOXES;
  float* candBuf  = wsf + WS_CAND;
  int*   gcnt     = (int*)(wsf + WS_GCNT);
  int*   labels_i = (int*)(wsf + WS_LAB);

  kA_softmax_decode<<<BQ * NP, 128, 0, stream>>>(logits, reg, props,
                                                 scoresW, boxesW, gcnt);
  kB_nms<<<BQ * CLS, 256, SMEM_B, stream>>>(scoresW, boxesW, candBuf, gcnt);
  kC_topk<<<BQ, 256, 0, stream>>>(candBuf, gcnt, out, labels_i);
  kD_mask<<<BQ * NKEEP, 256, 0, stream>>>(mlog, labels_i, out);
}